// DeformAlignBlock_14877766713887
// MI455X (gfx1250) — compile-verified
//
#include <hip/hip_runtime.h>
#include <hip/hip_bf16.h>

typedef float v2f __attribute__((ext_vector_type(2)));
typedef float v8f __attribute__((ext_vector_type(8)));

namespace {
constexpr int KK   = 9;     // 3x3 taps
constexpr int PADp = 1;
constexpr int Bn   = 4;
constexpr int Hh   = 96;
constexpr int Ww   = 96;
constexpr int HW   = Hh * Ww;
constexpr int OffC = 2 * KK; // 18 offset channels
}

// ---------------------------------------------------------------------------
// Plain 3x3 conv (zero pad) producing the 18 offset channels.
// blockIdx.y carries (b, o) so the weight pointer is block-uniform ->
// weights come through the scalar path (s_load / constant cache), VALU does
// only the FMAs; input reads are coalesced along x.
// ---------------------------------------------------------------------------
__global__ void conv3x3_offsets(const float* __restrict__ in,
                                const float* __restrict__ w,
                                const float* __restrict__ bias,
                                float* __restrict__ out,
                                int Cin) {
    const int o = blockIdx.y % OffC;
    const int b = blockIdx.y / OffC;
    const int p = blockIdx.x * blockDim.x + threadIdx.x;   // pixel in plane
    if (p >= HW) return;
    const int x = p % Ww;
    const int y = p / Ww;

    float acc = bias[o];
    const float* wp = w + (size_t)o * Cin * KK;            // block-uniform
    const float* ib = in + (size_t)b * Cin * HW;
    for (int c = 0; c < Cin; ++c) {
        const float* ip = ib + (size_t)c * HW;
        const float* wc = wp + c * KK;
#pragma unroll
        for (int ky = 0; ky < 3; ++ky) {
            int yy = y + ky - PADp;
            bool vy = (yy >= 0) && (yy < Hh);
#pragma unroll
            for (int kx = 0; kx < 3; ++kx) {
                int xx = x + kx - PADp;
                bool v = vy && (xx >= 0) && (xx < Ww);
                float iv = v ? ip[yy * Ww + xx] : 0.0f;
                acc = fmaf(iv, wc[ky * 3 + kx], acc);
            }
        }
    }
    out[((size_t)blockIdx.y * HW) + p] = acc;
}

// ---------------------------------------------------------------------------
// Pack deform weights for coalesced WMMA A-fragment loads:
//   wt[k][c/2][o] = float2( wd[o][c][k], wd[o][c+1][k] )
// so an A-fragment (channel pair c0+2*kh, out-channel o=t*16+ln) is one
// contiguous global_load_b64 with consecutive lanes at consecutive addresses.
// ---------------------------------------------------------------------------
template <int CIN, int O>
__global__ void pack_weights(const float* __restrict__ wd, float* __restrict__ wt) {
    int idx = blockIdx.x * blockDim.x + threadIdx.x;
    if (idx >= O * CIN * KK) return;
    int k = idx % KK;
    int c = (idx / KK) % CIN;
    int o = idx / (KK * CIN);
    wt[((((size_t)k * (CIN / 2) + (c >> 1)) * O + o) << 1) | (c & 1)] = wd[idx];
}

// ---------------------------------------------------------------------------
// Fused deformable conv as implicit GEMM on the f32 matrix pipe.
//   D[O x 16pix] += W[O x (C*9)] * S[(C*9) x 16pix]  via V_WMMA_F32_16X16X4_F32
// One wave per 16-pixel tile; every bilinear sample is gathered once and
// reused across all O output channels (NMT register accumulators).
// f32 WMMA operand layout (ISA 7.12.2): lane&15 indexes M (A) / N (B);
// lane>>4 selects the K-pair {0,1} vs {2,3}; A/B fragments are 2 VGPRs.
// The K-chunk loop is unrolled x4 so ~32 gathers + 24 weight b64 loads are in
// flight ahead of the dependent WMMAs (latency hiding at low occupancy); the
// f32 WMMAs co-execute with the VALU bilinear math of neighboring chunks.
// ---------------------------------------------------------------------------
template <int CIN, int O>
__global__ __launch_bounds__(128)
void deform_conv_wmma(const float* __restrict__ xin,   // [B, CIN, H, W]
                      const float* __restrict__ off,   // [B, 18,  H, W]
                      const float* __restrict__ wt,    // packed [9][CIN/2][O] float2
                      const float* __restrict__ bd,    // [O]
                      float* __restrict__ out) {       // [B, O,   H, W]
    constexpr int NMT = O / 16;          // out-channel tiles (4 or 6)
    constexpr int TPR = Ww / 16;         // pixel tiles per row (6)

    const int lane = threadIdx.x & 31;
    const int tile = blockIdx.x * (blockDim.x >> 5) + (threadIdx.x >> 5);
    if (tile >= Bn * Hh * TPR) return;   // wave-uniform; never taken (exact grid)

    const int xb = (tile % TPR) * 16;
    const int y  = (tile / TPR) % Hh;
    const int b  = tile / (TPR * Hh);
    const int ln = lane & 15;
    const int kh = lane >> 4;            // K-pair select within a 4-chunk
    const int px = xb + ln;              // this lane's pixel x

    v8f acc[NMT];
#pragma unroll
    for (int t = 0; t < NMT; ++t) acc[t] = v8f{};

    const float* xbase = xin + (size_t)b * CIN * HW;
    const v2f* wpk = (const v2f*)wt;

    for (int k = 0; k < KK; ++k) {
        const int ky = k / 3, kx = k % 3;
        // Bilinear sampling parameters for (pixel, tap k): computed once,
        // validity masks folded into the 4 corner weights.
        const float dy  = off[(((size_t)b * OffC + 2 * k + 0) * Hh + y) * Ww + px];
        const float dx  = off[(((size_t)b * OffC + 2 * k + 1) * Hh + y) * Ww + px];
        const float pyf = dy + (float)(y + ky - PADp);
        const float pxf = dx + (float)(px + kx - PADp);
        const float y0f = floorf(pyf);
        const float x0f = floorf(pxf);
        const float fy  = pyf - y0f;
        const float fx  = pxf - x0f;
        const int y0 = (int)y0f, x0 = (int)x0f;
        const int y1 = y0 + 1,  x1 = x0 + 1;
        const float my0 = (y0 >= 0 && y0 < Hh) ? 1.0f : 0.0f;
        const float my1 = (y1 >= 0 && y1 < Hh) ? 1.0f : 0.0f;
        const float mx0 = (x0 >= 0 && x0 < Ww) ? 1.0f : 0.0f;
        const float mx1 = (x1 >= 0 && x1 < Ww) ? 1.0f : 0.0f;
        const int y0c = min(max(y0, 0), Hh - 1);
        const int y1c = min(max(y1, 0), Hh - 1);
        const int x0c = min(max(x0, 0), Ww - 1);
        const int x1c = min(max(x1, 0), Ww - 1);
        const float w00 = (1.0f - fy) * (1.0f - fx) * my0 * mx0;
        const float w01 = (1.0f - fy) * fx          * my0 * mx1;
        const float w10 = fy          * (1.0f - fx) * my1 * mx0;
        const float w11 = fy          * fx          * my1 * mx1;
        const int i00 = y0c * Ww + x0c;
        const int i01 = y0c * Ww + x1c;
        const int i10 = y1c * Ww + x0c;
        const int i11 = y1c * Ww + x1c;

        const v2f* wk = wpk + (size_t)k * (CIN / 2) * O;

#pragma unroll 4
        for (int c0 = 0; c0 < CIN; c0 += 4) {
            // B fragment: bilinear samples for channels c0+2*kh {+0,+1}
            v2f sfrag;
#pragma unroll
            for (int j = 0; j < 2; ++j) {
                const float* cp = xbase + (size_t)(c0 + 2 * kh + j) * HW;
                sfrag[j] = fmaf(w00, cp[i00],
                           fmaf(w01, cp[i01],
                           fmaf(w10, cp[i10], w11 * cp[i11])));
            }
            // A fragments: one coalesced b64 load per out-channel tile.
            const v2f* wc = wk + ((c0 >> 1) + kh) * O;
#pragma unroll
            for (int t = 0; t < NMT; ++t) {
                v2f afrag = wc[t * 16 + ln];
                acc[t] = __builtin_amdgcn_wmma_f32_16x16x4_f32(
                    false, afrag, false, sfrag, (short)0, acc[t], false, false);
            }
        }
    }

    // D layout: VGPR v -> M = v + 8*kh (out channel), N = lane&15 (pixel).
    // Lanes 0..15 store 16 consecutive pixels per channel (coalesced).
#pragma unroll
    for (int t = 0; t < NMT; ++t) {
#pragma unroll
        for (int v = 0; v < 8; ++v) {
            const int o = t * 16 + v + 8 * kh;
            out[(((size_t)b * O + o) * Hh + y) * Ww + px] = acc[t][v] + bd[o];
        }
    }
}

// ---------------------------------------------------------------------------
extern "C" void kernel_launch(void* const* d_in, const int* in_sizes, int n_in,
                              void* d_out, int out_size, void* d_ws, size_t ws_size,
                              hipStream_t stream) {
    const float* x1  = (const float*)d_in[0];
    const float* y   = (const float*)d_in[1];
    const float* wo1 = (const float*)d_in[2];
    const float* bo1 = (const float*)d_in[3];
    const float* wd1 = (const float*)d_in[4];
    const float* bd1 = (const float*)d_in[5];
    const float* wo2 = (const float*)d_in[6];
    const float* bo2 = (const float*)d_in[7];
    const float* wd2 = (const float*)d_in[8];
    const float* bd2 = (const float*)d_in[9];
    const float* wo3 = (const float*)d_in[10];
    const float* bo3 = (const float*)d_in[11];
    const float* wd3 = (const float*)d_in[12];
    const float* bd3 = (const float*)d_in[13];
    float* outp = (float*)d_out;

    // Workspace layout (floats):
    //   off : B*18*HW        = 663,552   (2.7 MB)
    //   h1  : B*96*HW        = 3,538,944 (14.2 MB)
    //   h2  : B*96*HW        = 3,538,944 (14.2 MB)
    //   wT  : 96*96*9        = 82,944    (0.33 MB, reused per stage)
    float* off = (float*)d_ws;
    float* h1  = off + (size_t)Bn * OffC * HW;
    float* h2  = h1 + (size_t)Bn * 96 * HW;
    float* wT  = h2 + (size_t)Bn * 96 * HW;

    const dim3 offGrid((HW + 255) / 256, Bn * OffC);
    const int nTiles  = Bn * Hh * (Ww / 16);      // 2304 waves
    const int dBlocks = nTiles / 4;               // 4 wave32 per block

    // Stage 1: offsets from y (C=64), deform x1 (C=64 -> O=96)
    conv3x3_offsets<<<offGrid, 256, 0, stream>>>(y, wo1, bo1, off, 64);
    pack_weights<64, 96><<<(96 * 64 * KK + 255) / 256, 256, 0, stream>>>(wd1, wT);
    deform_conv_wmma<64, 96><<<dBlocks, 128, 0, stream>>>(x1, off, wT, bd1, h1);

    // Stage 2: offsets from h1 (C=96), deform h1 (C=96 -> O=96)
    conv3x3_offsets<<<offGrid, 256, 0, stream>>>(h1, wo2, bo2, off, 96);
    pack_weights<96, 96><<<(96 * 96 * KK + 255) / 256, 256, 0, stream>>>(wd2, wT);
    deform_conv_wmma<96, 96><<<dBlocks, 128, 0, stream>>>(h1, off, wT, bd2, h2);

    // Stage 3: offsets from h2 (C=96), deform h2 (C=96 -> O=64) -> d_out
    conv3x3_offsets<<<offGrid, 256, 0, stream>>>(h2, wo3, bo3, off, 96);
    pack_weights<96, 64><<<(64 * 96 * KK + 255) / 256, 256, 0, stream>>>(wd3, wT);
    deform_conv_wmma<96, 64><<<dBlocks, 128, 0, stream>>>(h2, off, wT, bd3, outp);
}